// dysOpt_52664888983702
// MI455X (gfx1250) — compile-verified
//
#include <hip/hip_runtime.h>
#include <hip/hip_bf16.h>
#include <stdint.h>

typedef __attribute__((ext_vector_type(2))) float v2f;
typedef __attribute__((ext_vector_type(8))) float v8f;

#define ALPHA    0.05f
#define MAX_ITER 200
#define TOL      0.01f
#define NN       64
#define CHUNKS   16   // 64 / 4 (K-chunks per WMMA_16x16x4 chain)

// Counter-based hash -> uniform [0,1)
__device__ __forceinline__ float u01(uint32_t x) {
    x ^= 61u ^ (x >> 16);
    x *= 9u;
    x ^= x >> 4;
    x *= 0x27d4eb2du;
    x ^= x >> 15;
    return (float)(x >> 8) * (1.0f / 16777216.0f);
}

// clip via single v_med3 (valid when lo <= hi, which holds for box bounds)
__device__ __forceinline__ float clip3(float z, float lo, float hi) {
    return __builtin_amdgcn_fmed3f(z, lo, hi);
}

// __launch_bounds__(256, 4): waves-per-eu >= 4 caps allocation at 256 VGPRs,
// keeping the hot loop out of the VGPR-MSB window (no s_set_vgpr_msb churn)
// and lifting occupancy; persistent state is ~200 VGPRs so no spills expected.
__global__ __launch_bounds__(256, 4) void dys_opt_kernel(
    const float* __restrict__ cost,   // (B, 64)
    const float* __restrict__ A,      // (1, 64)
    const float* __restrict__ b,      // (1,)
    const float* __restrict__ lb,     // (64,)
    const float* __restrict__ ub,     // (64,)
    const float* __restrict__ V,      // (64, 1)
    const float* __restrict__ UT,     // (1, 1)
    const float* __restrict__ s_inv,  // (1,)
    float* __restrict__ out,          // (B, 64)
    int batch)
{
    const int lane = (int)(threadIdx.x & 31u);
    const int wave = (int)((blockIdx.x * blockDim.x + threadIdx.x) >> 5);
    const int r    = lane & 15;   // row within 16-row tile (WMMA "M"/"N" index)
    const int h    = lane >> 4;   // which K-pair of each 4-chunk this lane holds

    if (wave * 16 >= batch) return;            // wave-uniform guard (EXEC stays all-ones)
    const int row = wave * 16 + r;

    const float b0   = b[0];
    const float utsi = UT[0] * s_inv[0];       // scalar collapse of UT^T and s_inv (M=1)
    const float a2   = ALPHA * ALPHA;
    const float C2   = 2.0f - a2;              // w  = C2*x - z - a*cost
    const float C1   = 1.0f - a2;              // z+ = C1*x - a*cost - coef*v

    // Per-lane register tiles: 32 elements of row `row`, as 16 float2 pairs.
    // Element k for chunk c: k = 4c + 2h, 4c + 2h + 1  (WMMA A- and B-matrix K layout).
    const size_t rowbase = (size_t)row * NN + (size_t)(2 * h);
    v2f ac[CHUNKS], zz[CHUNKS], lbv[CHUNKS], ubv[CHUNKS], av[CHUNKS], vs[CHUNKS];

#pragma unroll
    for (int c = 0; c < CHUNKS; ++c) {
        const int k = 4 * c + 2 * h;
        v2f cc = *(const v2f*)(cost + rowbase + 4 * c);
        ac[c].x = ALPHA * cc.x;                // pre-scaled alpha*cost
        ac[c].y = ALPHA * cc.y;
        lbv[c] = *(const v2f*)(lb + k);
        ubv[c] = *(const v2f*)(ub + k);
        av[c]  = *(const v2f*)(A + k);         // A row doubles as the WMMA A-operand
        v2f vk = *(const v2f*)(V + k);         // V[:,0]
        vs[c].x = utsi * vk.x;                 // pre-scaled so coef = dot - b0
        vs[c].y = utsi * vk.y;
    }

    // z0 = uniform * (ub_safe - lb_safe) + lb_safe
#pragma unroll
    for (int c = 0; c < CHUNKS; ++c) {
        const int k = 4 * c + 2 * h;
        float l0 = isfinite(lbv[c].x) ? lbv[c].x : 0.0f;
        float l1 = isfinite(lbv[c].y) ? lbv[c].y : 0.0f;
        float u0 = isfinite(ubv[c].x) ? ubv[c].x : 1.0f;
        float u1 = isfinite(ubv[c].y) ? ubv[c].y : 1.0f;
        zz[c].x = u01((uint32_t)(row * NN + k))     * (u0 - l0) + l0;
        zz[c].y = u01((uint32_t)(row * NN + k + 1)) * (u1 - l1) + l1;
    }

    const float tol2 = TOL * TOL;

    for (int it = 0; it < MAX_ITER; ++it) {
        // ---- Phase A: rowdot[n] = sum_k A0k * w[n][k] via chained V_WMMA_F32_16X16X4_F32.
        // A-operand carries the A-row (all 16 rows identical), B-operand carries w.
        // D[m][n] = weighted dot for tile-row n, replicated over m -> read acc[0].
        v8f acc = {};
#pragma unroll
        for (int c = 0; c < CHUNKS; ++c) {
            float x0 = clip3(zz[c].x, lbv[c].x, ubv[c].x);
            float x1 = clip3(zz[c].y, lbv[c].y, ubv[c].y);
            v2f w;
            w.x = fmaf(C2, x0, -zz[c].x) - ac[c].x;
            w.y = fmaf(C2, x1, -zz[c].y) - ac[c].y;
            acc = __builtin_amdgcn_wmma_f32_16x16x4_f32(
                false, av[c], false, w, (short)0, acc, false, false);
        }
        const float coef = acc[0] - b0;        // vs already folded UT*s_inv

        // ---- Phase B: z+ = C1*x - a*cost - coef*vs ; accumulate ||dz||^2 per lane.
        float nrm0 = 0.0f, nrm1 = 0.0f;
#pragma unroll
        for (int c = 0; c < CHUNKS; ++c) {
            float x0 = clip3(zz[c].x, lbv[c].x, ubv[c].x);
            float x1 = clip3(zz[c].y, lbv[c].y, ubv[c].y);
            float zn0 = fmaf(C1, x0, fmaf(-coef, vs[c].x, -ac[c].x));
            float zn1 = fmaf(C1, x1, fmaf(-coef, vs[c].y, -ac[c].y));
            float d0 = zn0 - zz[c].x, d1 = zn1 - zz[c].y;
            nrm0 = fmaf(d0, d0, nrm0);
            nrm1 = fmaf(d1, d1, nrm1);
            zz[c].x = zn0; zz[c].y = zn1;
        }
        float nrm = nrm0 + nrm1;
        // Row norm = this lane's half + partner half (lane ^ 16 holds the other K-slots).
        nrm += __shfl_xor(nrm, 16, 32);
        if (!__any(nrm > tol2)) break;         // wave-uniform exit: EXEC stays all-ones
    }

    // ---- One differentiable DYS (JFB) step, then final box projection.
    {
        v8f acc = {};
#pragma unroll
        for (int c = 0; c < CHUNKS; ++c) {
            float x0 = clip3(zz[c].x, lbv[c].x, ubv[c].x);
            float x1 = clip3(zz[c].y, lbv[c].y, ubv[c].y);
            v2f w;
            w.x = fmaf(C2, x0, -zz[c].x) - ac[c].x;
            w.y = fmaf(C2, x1, -zz[c].y) - ac[c].y;
            acc = __builtin_amdgcn_wmma_f32_16x16x4_f32(
                false, av[c], false, w, (short)0, acc, false, false);
        }
        const float coef = acc[0] - b0;
#pragma unroll
        for (int c = 0; c < CHUNKS; ++c) {
            float x0 = clip3(zz[c].x, lbv[c].x, ubv[c].x);
            float x1 = clip3(zz[c].y, lbv[c].y, ubv[c].y);
            float zn0 = fmaf(C1, x0, fmaf(-coef, vs[c].x, -ac[c].x));
            float zn1 = fmaf(C1, x1, fmaf(-coef, vs[c].y, -ac[c].y));
            v2f sol;
            sol.x = clip3(zn0, lbv[c].x, ubv[c].x);
            sol.y = clip3(zn1, lbv[c].y, ubv[c].y);
            *(v2f*)(out + rowbase + 4 * c) = sol;
        }
    }
}

extern "C" void kernel_launch(void* const* d_in, const int* in_sizes, int n_in,
                              void* d_out, int out_size, void* d_ws, size_t ws_size,
                              hipStream_t stream) {
    const float* cost  = (const float*)d_in[0];
    const float* A     = (const float*)d_in[1];
    const float* b     = (const float*)d_in[2];
    const float* lb    = (const float*)d_in[3];
    const float* ub    = (const float*)d_in[4];
    const float* V     = (const float*)d_in[5];
    const float* UT    = (const float*)d_in[6];
    const float* s_inv = (const float*)d_in[7];
    float* out = (float*)d_out;

    const int batch  = in_sizes[0] / NN;          // 262144
    const int waves  = (batch + 15) / 16;         // one wave32 per 16 rows
    const int blocks = (waves + 7) / 8;           // 8 waves (256 threads) per block
    dys_opt_kernel<<<blocks, 256, 0, stream>>>(cost, A, b, lb, ub, V, UT, s_inv,
                                               out, batch);
}